// GCNConvMultiEdgeset_25340307046680
// MI455X (gfx1250) — compile-verified
//
#include <hip/hip_runtime.h>
#include <hip/hip_bf16.h>

#define NN 10000
#define NE 640000
#define C  128
#define BF 16

typedef __attribute__((ext_vector_type(2))) float v2f;
typedef __attribute__((ext_vector_type(8))) float v8f;

__device__ __forceinline__ float gelu_exact(float v) {
    // 0.5*v*(1+erf(v/sqrt(2))) -- exact (non-tanh) GELU, matches jax approximate=False
    return 0.5f * v * (1.0f + erff(v * 0.70710678118654752440f));
}

// ---------------- Kernel 1: init counters (self-loop contributes 1) + zero accumulator
__global__ void __launch_bounds__(256) k_init(unsigned* cntRow, unsigned* cntCol, float* accum) {
    int i = blockIdx.x * blockDim.x + threadIdx.x;
    if (i < NN * C) accum[i] = 0.0f;
    if (i < NN) { cntRow[i] = 1u; cntCol[i] = 1u; }
}

// ---------------- Kernel 2: integer degree counting over edges
__global__ void __launch_bounds__(256) k_degree(const int* __restrict__ ei,
                                                unsigned* cntRow, unsigned* cntCol) {
    int e = blockIdx.x * blockDim.x + threadIdx.x;
    if (e < NE) {
        atomicAdd(&cntRow[ei[e]], 1u);        // row indices
        atomicAdd(&cntCol[ei[NE + e]], 1u);   // col indices
    }
}

// ---------------- Kernel 3: deg -> 1/sqrt(deg)  (deg >= 1 always, no zero case)
__global__ void __launch_bounds__(256) k_invsqrt(const unsigned* __restrict__ cntRow,
                                                 const unsigned* __restrict__ cntCol,
                                                 float* invRow, float* invCol) {
    int i = blockIdx.x * blockDim.x + threadIdx.x;
    if (i < NN) {
        invRow[i] = rsqrtf((float)cntRow[i]);
        invCol[i] = rsqrtf((float)cntCol[i]);
    }
}

// ---------------- Kernel 4: per-edge bond-encoder WMMA + GELU + scatter-add
// One wave handles 16 edges. emb tile = [16 edges x 128 ch] via 8 N-tiles x 4 K-chunks
// of v_wmma_f32_16x16x4_f32 (f32 A/B/C -- bit-faithful to the f32 reference GEMM).
__global__ void __launch_bounds__(256) k_edge_msg(
    const float* __restrict__ x, const float* __restrict__ edge_attr,
    const float* __restrict__ edge_weight, const float* __restrict__ W_bond,
    const float* __restrict__ b_bond, const int* __restrict__ ei,
    const float* __restrict__ invRow, const float* __restrict__ invCol,
    float* __restrict__ accum)
{
    __shared__ int   sRow[8][16];
    __shared__ int   sCol[8][16];
    __shared__ float sScale[8][16];

    const int wave  = threadIdx.x >> 5;
    const int lane  = threadIdx.x & 31;
    const int eBase = (blockIdx.x * 8 + wave) * 16;

    if (lane < 16) {
        int e = eBase + lane;
        int r = ei[e];
        int c = ei[NE + e];
        sRow[wave][lane] = r;
        sCol[wave][lane] = c;
        sScale[wave][lane] = invRow[r] * invCol[c] * edge_weight[e];
    }
    __syncthreads();

    const int n     = lane & 15;              // N for B/C, M for A loads
    const int kHalf = (lane >> 4) ? 2 : 0;    // A/B f32 layout: lanes 16-31 hold K+2
    const int mBase = (lane >> 4) ? 8 : 0;    // C/D layout: lanes 16-31 hold M+8

    // A fragments: edge_attr tile [16 edges x 16 bond feats], 4 K-chunks of 4
    v2f aF[4];
#pragma unroll
    for (int kc = 0; kc < 4; ++kc) {
        const float* ap = edge_attr + (size_t)(eBase + n) * BF + kc * 4 + kHalf;
        aF[kc].x = ap[0];   // K = 4*kc + kHalf
        aF[kc].y = ap[1];   // K = 4*kc + kHalf + 1
    }

#pragma unroll
    for (int nt = 0; nt < 8; ++nt) {
        v8f acc = {0.f,0.f,0.f,0.f,0.f,0.f,0.f,0.f};
#pragma unroll
        for (int kc = 0; kc < 4; ++kc) {
            int ko = kc * 4 + kHalf;
            v2f b;
            b.x = W_bond[ ko      * C + nt * 16 + n];
            b.y = W_bond[(ko + 1) * C + nt * 16 + n];
            acc = __builtin_amdgcn_wmma_f32_16x16x4_f32(
                      false, aF[kc], false, b, (short)0, acc, false, false);
        }
#pragma unroll
        for (int r = 0; r < 8; ++r) {
            int   m    = mBase + r;          // edge within the 16-edge tile
            int   ch   = nt * 16 + n;        // output channel
            int   rowm = sRow[wave][m];
            int   colm = sCol[wave][m];
            float sc   = sScale[wave][m];
            float v = acc[r] + b_bond[ch] + x[(size_t)rowm * C + ch];  // coalesced L2 gather
            unsafeAtomicAdd(&accum[(size_t)colm * C + ch], gelu_exact(v) * sc);
        }
    }
}

// ---------------- Kernel 5: H = accum + selfloop-gelu term; out = H @ W_lin + b_lin
// One block per 16-row tile; H staged in LDS; 8 waves x (1 col-tile each) x 32 f32 WMMAs.
__global__ void __launch_bounds__(256) k_out_gemm(
    const float* __restrict__ x, const float* __restrict__ accum,
    const float* __restrict__ invRow, const float* __restrict__ invCol,
    const float* __restrict__ W_lin, const float* __restrict__ b_lin,
    float* __restrict__ out)
{
    __shared__ float hTile[16][C];    // 8 KB
    const int tid     = threadIdx.x;
    const int rowBase = blockIdx.x * 16;

#pragma unroll
    for (int k = 0; k < 8; ++k) {               // 256 thr * 8 = 2048 = 16*128
        int idx = tid + k * 256;
        int rr  = idx >> 7;
        int cc  = idx & 127;
        int i   = rowBase + rr;
        float xv = x[(size_t)i * C + cc];
        hTile[rr][cc] = accum[(size_t)i * C + cc]
                      + gelu_exact(xv) * invRow[i] * invCol[i];  // self-loop message
    }
    __syncthreads();

    const int wave  = tid >> 5;                 // col-tile 0..7
    const int lane  = tid & 31;
    const int n     = lane & 15;
    const int kHalf = (lane >> 4) ? 2 : 0;
    const int mBase = (lane >> 4) ? 8 : 0;

    v8f acc = {0.f,0.f,0.f,0.f,0.f,0.f,0.f,0.f};
#pragma unroll
    for (int kc = 0; kc < 32; ++kc) {           // K = 128 in chunks of 4
        int ko = kc * 4 + kHalf;
        v2f a, b;
        a.x = hTile[n][ko];
        a.y = hTile[n][ko + 1];
        b.x = W_lin[ ko      * C + wave * 16 + n];
        b.y = W_lin[(ko + 1) * C + wave * 16 + n];
        acc = __builtin_amdgcn_wmma_f32_16x16x4_f32(
                  false, a, false, b, (short)0, acc, false, false);
    }
#pragma unroll
    for (int r = 0; r < 8; ++r) {
        int m  = mBase + r;
        int ch = wave * 16 + n;
        out[(size_t)(rowBase + m) * C + ch] = acc[r] + b_lin[ch];
    }
}

extern "C" void kernel_launch(void* const* d_in, const int* in_sizes, int n_in,
                              void* d_out, int out_size, void* d_ws, size_t ws_size,
                              hipStream_t stream) {
    const float* x           = (const float*)d_in[0];
    const float* edge_attr   = (const float*)d_in[1];
    const float* edge_weight = (const float*)d_in[2];
    const float* W_bond      = (const float*)d_in[3];
    const float* b_bond      = (const float*)d_in[4];
    const float* W_lin       = (const float*)d_in[5];
    const float* b_lin       = (const float*)d_in[6];
    const int*   ei          = (const int*)d_in[7];

    unsigned* cntRow = (unsigned*)d_ws;                // NN
    unsigned* cntCol = cntRow + NN;                    // NN
    float*    invRow = (float*)(cntCol + NN);          // NN
    float*    invCol = invRow + NN;                    // NN
    float*    accum  = invCol + NN;                    // NN*C  (~5.1 MB total ws use)
    float*    out    = (float*)d_out;

    k_init   <<<(NN * C + 255) / 256, 256, 0, stream>>>(cntRow, cntCol, accum);
    k_degree <<<(NE + 255) / 256,     256, 0, stream>>>(ei, cntRow, cntCol);
    k_invsqrt<<<(NN + 255) / 256,     256, 0, stream>>>(cntRow, cntCol, invRow, invCol);
    k_edge_msg<<<NE / 128, 256, 0, stream>>>(x, edge_attr, edge_weight, W_bond, b_bond,
                                             ei, invRow, invCol, accum);
    k_out_gemm<<<NN / 16, 256, 0, stream>>>(x, accum, invRow, invCol, W_lin, b_lin, out);
}